// AttentionFusion_18897856102951
// MI455X (gfx1250) — compile-verified
//
#include <hip/hip_runtime.h>
#include <cstdint>

typedef float v2f __attribute__((ext_vector_type(2)));
typedef float v8f __attribute__((ext_vector_type(8)));

#define BLOCK 256
#define DCOLS 2048

__global__ __launch_bounds__(BLOCK) void attn_fuse_kernel(
    const float* __restrict__ x1,
    const float* __restrict__ x2,
    const float* __restrict__ w,
    float* __restrict__ fused,
    float* __restrict__ alpha_out,
    int D)
{
    __shared__ float lds_x1[DCOLS];
    __shared__ float lds_x2[DCOLS];
    __shared__ float red1[8];
    __shared__ float red2[8];
    __shared__ float s_alpha;

    const int t    = threadIdx.x;
    const int row  = blockIdx.x;
    const int half = D >> 1;                      // 1024

    // ---- Stage 1: async copy both rows into LDS (CDNA5 async-to-LDS path) ----
    const uint64_t gbase1 = (uint64_t)(uintptr_t)(x1 + (size_t)row * D);
    const uint64_t gbase2 = (uint64_t)(uintptr_t)(x2 + (size_t)row * D);

    // byte offsets of this thread's two 16B chunks (512B contiguous per wave per op)
    const uint32_t off0 = (uint32_t)(t * 16);                 // floats [4t, 4t+4)
    const uint32_t off1 = (uint32_t)(half * 4 + t * 16);      // floats [half+4t, ...)

    // flat 'shared' pointers: low 32 bits == workgroup-relative LDS byte address
    const uint32_t l1 = (uint32_t)(uintptr_t)lds_x1;
    const uint32_t l2 = (uint32_t)(uintptr_t)lds_x2;
    const uint32_t l1a = l1 + off0, l1b = l1 + off1;
    const uint32_t l2a = l2 + off0, l2b = l2 + off1;

    asm volatile(
        "global_load_async_to_lds_b128 %0, %4, %6\n\t"
        "global_load_async_to_lds_b128 %1, %5, %6\n\t"
        "global_load_async_to_lds_b128 %2, %4, %7\n\t"
        "global_load_async_to_lds_b128 %3, %5, %7\n\t"
        :
        : "v"(l1a), "v"(l1b), "v"(l2a), "v"(l2b),
          "v"(off0), "v"(off1),
          "s"(gbase1), "s"(gbase2)
        : "memory");
    asm volatile("s_wait_asynccnt 0" ::: "memory");
    __syncthreads();

    // ---- Stage 2: per-thread partial dots; keep row data in registers ----
    const float4* w4 = (const float4*)w;
    const float4 wA = w4[t];
    const float4 wB = w4[(half >> 2) + t];

    const float4 a1 = *(const float4*)(lds_x1 + 4 * t);
    const float4 b1 = *(const float4*)(lds_x1 + half + 4 * t);
    const float4 a2 = *(const float4*)(lds_x2 + 4 * t);
    const float4 b2 = *(const float4*)(lds_x2 + half + 4 * t);

    float p1 = a1.x * wA.x + a1.y * wA.y + a1.z * wA.z + a1.w * wA.w
             + b1.x * wB.x + b1.y * wB.y + b1.z * wB.z + b1.w * wB.w;
    float p2 = a2.x * wA.x + a2.y * wA.y + a2.z * wA.z + a2.w * wA.w
             + b2.x * wB.x + b2.y * wB.y + b2.z * wB.z + b2.w * wB.w;

    // ---- Stage 3: wave32 shuffle reduction, then one WMMA for cross-wave sum ----
    #pragma unroll
    for (int off = 16; off >= 1; off >>= 1) {
        p1 += __shfl_xor(p1, off, 32);
        p2 += __shfl_xor(p2, off, 32);
    }
    const int lane = t & 31;
    const int wv   = t >> 5;
    if (lane == 0) { red1[wv] = p1; red2[wv] = p2; }
    __syncthreads();

    // A (16x4 f32): row 0 holds the 4 paired s1 partials, row 1 the s2 partials.
    // Layout: lanes 0-15 supply (M=lane, K=0/1) via a.x/a.y; lanes 16-31 supply K=2/3.
    // B = ones(4x16)  =>  D[i][j] = sum_k A[i][k]  => D row 0 = s1, row 1 = s2,
    // broadcast across all 16 N-columns for free. Executed by ALL waves (EXEC all-1s).
    v2f a; a.x = 0.0f; a.y = 0.0f;
    if (t == 0)  { a.x = red1[0] + red1[1]; a.y = red1[2] + red1[3]; }
    if (t == 16) { a.x = red1[4] + red1[5]; a.y = red1[6] + red1[7]; }
    if (t == 1)  { a.x = red2[0] + red2[1]; a.y = red2[2] + red2[3]; }
    if (t == 17) { a.x = red2[4] + red2[5]; a.y = red2[6] + red2[7]; }
    v2f b; b.x = 1.0f; b.y = 1.0f;
    v8f c = {};
    v8f d = __builtin_amdgcn_wmma_f32_16x16x4_f32(
        /*neg_a=*/false, a, /*neg_b=*/false, b,
        /*c_mod=*/(short)0, c, /*reuse_a=*/false, /*reuse_b=*/false);

    if (t == 0) {
        const float s1 = d[0];   // M=0, lane 0
        const float s2 = d[1];   // M=1, lane 0
        const float al = 1.0f / (1.0f + __expf(s2 - s1));  // sigmoid(s1 - s2)
        s_alpha = al;
        alpha_out[2 * (size_t)row]     = al;
        alpha_out[2 * (size_t)row + 1] = 1.0f - al;
    }
    __syncthreads();

    // ---- Stage 4: blend from registers, coalesced b128 stores ----
    const float al = s_alpha;
    float4 o0, o1;
    o0.x = fmaf(al, a1.x - a2.x, a2.x);
    o0.y = fmaf(al, a1.y - a2.y, a2.y);
    o0.z = fmaf(al, a1.z - a2.z, a2.z);
    o0.w = fmaf(al, a1.w - a2.w, a2.w);
    o1.x = fmaf(al, b1.x - b2.x, b2.x);
    o1.y = fmaf(al, b1.y - b2.y, b2.y);
    o1.z = fmaf(al, b1.z - b2.z, b2.z);
    o1.w = fmaf(al, b1.w - b2.w, b2.w);

    float4* out4 = (float4*)(fused + (size_t)row * D);
    out4[t] = o0;
    out4[(half >> 2) + t] = o1;
}

extern "C" void kernel_launch(void* const* d_in, const int* in_sizes, int n_in,
                              void* d_out, int out_size, void* d_ws, size_t ws_size,
                              hipStream_t stream) {
    const float* x1 = (const float*)d_in[0];
    const float* x2 = (const float*)d_in[1];
    const float* w  = (const float*)d_in[2];
    float* out = (float*)d_out;

    const int D = in_sizes[2];            // 2048
    const int N = in_sizes[0] / D;        // 16384

    float* fused = out;                   // [N, D]
    float* alpha = out + (size_t)N * D;   // [N, 2]

    attn_fuse_kernel<<<N, BLOCK, 0, stream>>>(x1, x2, w, fused, alpha, D);
}